// PromptParser_68375879352840
// MI455X (gfx1250) — compile-verified
//
#include <hip/hip_runtime.h>
#include <math.h>

typedef __attribute__((ext_vector_type(2))) float v2f;
typedef __attribute__((ext_vector_type(8))) float v8f;

namespace {
constexpr int kB   = 16;
constexpr int kN   = 64;
constexpr int kC   = 256;
constexpr int kBNC = kB * kN * kC;   // 262144
constexpr int kCX  = kC * kC;        // 65536
constexpr int kHID = 12;
}

// ---------------------------------------------------------------------------
// Kernel 1: per-element 3->12->1 MLP (exact GELU) producing etpp, plus
// per-(b,n) sign-split sums of s^2 where s = tmp + p1 + p2.
__global__ void k_mlp_reduce(const float* __restrict__ tmp,
                             const float* __restrict__ p1,
                             const float* __restrict__ p2,
                             const float* __restrict__ w1,
                             const float* __restrict__ b1,
                             const float* __restrict__ w2,
                             const float* __restrict__ b2,
                             float* __restrict__ etpp,
                             float* __restrict__ Ppos,
                             float* __restrict__ Pneg) {
  int bn  = blockIdx.x;        // b*kN + n
  int c   = threadIdx.x;       // 0..255
  int idx = bn * kC + c;

  float t  = tmp[idx];
  float q1 = p1[idx];
  float q2 = p2[idx];

  float acc = b2[0];
#pragma unroll
  for (int h = 0; h < kHID; ++h) {
    float z = fmaf(t, w1[h], fmaf(q1, w1[kHID + h], fmaf(q2, w1[2 * kHID + h], b1[h])));
    float g = 0.5f * z * (1.0f + erff(z * 0.70710678118654752f));  // exact GELU
    acc = fmaf(g, w2[h], acc);
  }
  etpp[idx] = acc;

  float sv = t + q1 + q2;
  float sq = sv * sv;

  __shared__ float rp[kC];
  __shared__ float rn[kC];
  rp[c] = (sv >= 0.0f) ? sq : 0.0f;
  rn[c] = (sv >= 0.0f) ? 0.0f : sq;
  __syncthreads();
  for (int off = kC / 2; off > 0; off >>= 1) {
    if (c < off) { rp[c] += rp[c + off]; rn[c] += rn[c + off]; }
    __syncthreads();
  }
  if (c == 0) { Ppos[bn] = rp[0]; Pneg[bn] = rn[0]; }
}

// ---------------------------------------------------------------------------
// Kernel 2: S[b,c,x] = sum_n etpp[b,n,c] * image[b,n,x]
// Per batch: (256x64) @ (64x256) GEMM via V_WMMA_F32_16X16X4_F32.
// One wave32 per 16x16 output tile; 256 tiles/batch, 16 batches = 4096 waves.
__global__ void k_gemm1_wmma(const float* __restrict__ etpp,
                             const float* __restrict__ image,
                             float* __restrict__ S) {
  int gtid = blockIdx.x * blockDim.x + threadIdx.x;
  int wave = gtid >> 5;
  int lane = gtid & 31;
  int b    = wave >> 8;              // 256 tiles per batch
  int tile = wave & 255;
  int c0   = (tile >> 4) << 4;
  int x0   = (tile & 15) << 4;
  int lh   = lane & 15;              // M / N index within tile
  int hi   = lane >> 4;              // selects K pair {0,1} vs {2,3}

  const float* Eb = etpp  + b * kN * kC;
  const float* Ib = image + b * kN * kC;

  v8f acc = {};
#pragma unroll
  for (int k0 = 0; k0 < kN; k0 += 4) {
    int k = k0 + (hi << 1);
    v2f a, bf;
    a[0]  = Eb[(k + 0) * kC + (c0 + lh)];   // A[m=c][k=n] = etpp[b,n,c]
    a[1]  = Eb[(k + 1) * kC + (c0 + lh)];
    bf[0] = Ib[(k + 0) * kC + (x0 + lh)];   // B[k=n][j=x] = image[b,n,x]
    bf[1] = Ib[(k + 1) * kC + (x0 + lh)];
    acc = __builtin_amdgcn_wmma_f32_16x16x4_f32(false, a, false, bf,
                                                (short)0, acc, false, false);
  }

  float* Sb = S + b * kCX;
#pragma unroll
  for (int r = 0; r < 8; ++r) {
    Sb[(c0 + r + 8 * hi) * kC + (x0 + lh)] = acc[r];
  }
}

// ---------------------------------------------------------------------------
// Kernel 3: 3x3 conv (shared kernel, zero pad 1) + bias, per batch 256x256.
__global__ void k_conv3x3(const float* __restrict__ S,
                          const float* __restrict__ gw,
                          const float* __restrict__ gb,
                          float* __restrict__ A) {
  int idx = blockIdx.x * blockDim.x + threadIdx.x;  // b*65536 + c*256 + x
  int b = idx >> 16;
  int c = (idx >> 8) & 255;
  int x = idx & 255;
  const float* Sb = S + b * kCX;
  float acc = gb[0];
#pragma unroll
  for (int dy = -1; dy <= 1; ++dy) {
    int cc = c + dy;
    if (cc < 0 || cc >= kC) continue;
#pragma unroll
    for (int dx = -1; dx <= 1; ++dx) {
      int xx = x + dx;
      if (xx < 0 || xx >= kC) continue;
      acc = fmaf(Sb[cc * kC + xx], gw[(dy + 1) * 3 + (dx + 1)], acc);
    }
  }
  A[idx] = acc;
}

// ---------------------------------------------------------------------------
// Kernel 4: U[b,n,c] = sum_x A[b,c,x] * image[b,n,x]
// Per batch: (256x256) @ (256x64) GEMM via V_WMMA_F32_16X16X4_F32.
// 64 tiles/batch (16 c-tiles x 4 n-tiles), 16 batches = 1024 waves.
// A and B fragments are K-contiguous -> 8-byte vector loads.
__global__ void k_gemm2_wmma(const float* __restrict__ A,
                             const float* __restrict__ image,
                             float* __restrict__ U) {
  int gtid = blockIdx.x * blockDim.x + threadIdx.x;
  int wave = gtid >> 5;
  int lane = gtid & 31;
  int b    = wave >> 6;              // 64 tiles per batch
  int tile = wave & 63;
  int c0   = (tile >> 2) << 4;
  int n0   = (tile & 3) << 4;
  int lh   = lane & 15;
  int hi   = lane >> 4;

  const float* arow = A     + b * kCX      + (c0 + lh) * kC;  // A[c0+lh][x]
  const float* brow = image + b * kN * kC  + (n0 + lh) * kC;  // B[x][n0+lh] = image[b,n0+lh,x]

  v8f acc = {};
#pragma unroll 8
  for (int k0 = 0; k0 < kC; k0 += 4) {
    int k = k0 + (hi << 1);
    v2f a  = *(const v2f*)(arow + k);
    v2f bf = *(const v2f*)(brow + k);
    acc = __builtin_amdgcn_wmma_f32_16x16x4_f32(false, a, false, bf,
                                                (short)0, acc, false, false);
  }

  float* Ub = U + b * kN * kC;
#pragma unroll
  for (int r = 0; r < 8; ++r) {
    Ub[(n0 + lh) * kC + (c0 + r + 8 * hi)] = acc[r];
  }
}

// ---------------------------------------------------------------------------
// Kernel 5: res[b,n,c] = max(u,g)*Ppos[b,n] + min(u,g)*Pneg[b,n]; g = image.
// Also copies image_embedding into the first output slot (tuple output).
__global__ void k_finalize(const float* __restrict__ image,
                           const float* __restrict__ U,
                           const float* __restrict__ Ppos,
                           const float* __restrict__ Pneg,
                           float* __restrict__ out) {
  int idx = blockIdx.x * blockDim.x + threadIdx.x;
  int bn  = idx >> 8;
  float g = image[idx];
  float u = U[idx];
  float mx = fmaxf(u, g);
  float mn = fminf(u, g);
  out[idx]        = g;                               // output 0: image_embedding
  out[kBNC + idx] = fmaf(mx, Ppos[bn], mn * Pneg[bn]);  // output 1: res
}

// ---------------------------------------------------------------------------
extern "C" void kernel_launch(void* const* d_in, const int* in_sizes, int n_in,
                              void* d_out, int out_size, void* d_ws, size_t ws_size,
                              hipStream_t stream) {
  (void)in_sizes; (void)n_in; (void)out_size; (void)ws_size;

  const float* image = (const float*)d_in[0];
  const float* tmp   = (const float*)d_in[1];
  const float* p1    = (const float*)d_in[2];
  const float* p2    = (const float*)d_in[3];
  const float* w1    = (const float*)d_in[4];   // (3,12) row-major
  const float* b1    = (const float*)d_in[5];   // (12,)
  const float* w2    = (const float*)d_in[6];   // (12,1)
  const float* b2    = (const float*)d_in[7];   // (1,)
  const float* gw    = (const float*)d_in[8];   // (3,3) row-major
  const float* gb    = (const float*)d_in[9];   // (1,)
  float* out = (float*)d_out;

  // Workspace layout (floats): etpp | Ppos | Pneg | S | Aconv | U  (~10.5 MB)
  float* ws   = (float*)d_ws;
  float* etpp = ws;                        // kBNC
  float* Ppos = etpp + kBNC;               // kB*kN
  float* Pneg = Ppos + kB * kN;            // kB*kN
  float* Smat = Pneg + kB * kN;            // kB*kCX
  float* Amat = Smat + kB * kCX;           // kB*kCX
  float* Umat = Amat + kB * kCX;           // kBNC

  k_mlp_reduce<<<kB * kN, kC, 0, stream>>>(tmp, p1, p2, w1, b1, w2, b2,
                                           etpp, Ppos, Pneg);
  // 4096 waves * 32 lanes / 256 threads = 512 blocks
  k_gemm1_wmma<<<512, 256, 0, stream>>>(etpp, image, Smat);
  k_conv3x3<<<(kB * kCX) / 256, 256, 0, stream>>>(Smat, gw, gb, Amat);
  // 1024 waves * 32 lanes / 256 threads = 128 blocks
  k_gemm2_wmma<<<128, 256, 0, stream>>>(Amat, image, Umat);
  k_finalize<<<kBNC / 256, 256, 0, stream>>>(image, Umat, Ppos, Pneg, out);
}